// PartialEncoder_25984552141001
// MI455X (gfx1250) — compile-verified
//
#include <hip/hip_runtime.h>
#include <hip/hip_bf16.h>

typedef __attribute__((ext_vector_type(2))) float f32x2;
typedef __attribute__((ext_vector_type(8))) float f32x8;

#define LN_EPS   1e-5f
#define MEAN_EPS 1e-8f
#define HDIM 128
#define CDIM 32
#define EDIM 128
#define NCAT 4
#define CHUNK_D 1280
#define TILES_PER_CHUNK (CHUNK_D / 16)
#define NWAVES 4
#define BLOCK_T (NWAVES * 32)
#define TILE_FLOATS (16 * HDIM)   // one 16-row tile of epg: 8 KB

// Raw lane-data-share ops (EXEC is all-ones wherever these are used):
//  ds_swizzle group-of-32 encoding: offset = xor_mask<<10 | or_mask<<5 | and_mask
#define SWZ_XOR(v, XM) \
    __int_as_float(__builtin_amdgcn_ds_swizzle(__float_as_int(v), (((XM) << 10) | 0x1F)))
#define BPERM(idx, v) \
    __int_as_float(__builtin_amdgcn_ds_bpermute((idx), __float_as_int(v)))

// ---------------------------------------------------------------------------
// Kernel 1: per-d precompute.
//   epg[d][h] = ((fe[d] @ h_W1[1:])[h] + h_b1[h]) * ln1_g[h]   (gain pre-folded)
//   mv/vv/qv  = mean / var of raw ep row, cov(u, ep row)
//   scal      = mean(u), var(u)
// LN1 stats of h1 = a*u + ep are then closed-form in a = x[b,d].
// ---------------------------------------------------------------------------
__global__ void pe_precompute(const float* __restrict__ fe,
                              const float* __restrict__ hW1,
                              const float* __restrict__ hb1,
                              const float* __restrict__ ln1g,
                              float* __restrict__ epg, float* __restrict__ mv,
                              float* __restrict__ qv, float* __restrict__ vv,
                              float* __restrict__ scal, int D) {
    int d = blockIdx.x * blockDim.x + threadIdx.x;
    if (d >= D) return;
    float su = 0.f, suu = 0.f;
    #pragma unroll 8
    for (int h = 0; h < HDIM; ++h) { float uh = hW1[h]; su += uh; suu = fmaf(uh, uh, suu); }
    const float mu_u  = su  * (1.f / HDIM);
    const float var_u = suu * (1.f / HDIM) - mu_u * mu_u;
    if (d == 0) { scal[0] = mu_u; scal[1] = var_u; }

    float fr[CDIM];
    #pragma unroll
    for (int c = 0; c < CDIM; ++c) fr[c] = fe[(size_t)d * CDIM + c];

    float sv = 0.f, svv = 0.f, suv = 0.f;
    float* eprow = epg + (size_t)d * HDIM;
    for (int h = 0; h < HDIM; ++h) {
        float s = hb1[h];
        #pragma unroll
        for (int c = 0; c < CDIM; ++c) s = fmaf(fr[c], hW1[(1 + c) * HDIM + h], s);
        eprow[h] = s * ln1g[h];                  // store with gain folded in
        sv += s; svv = fmaf(s, s, svv); suv = fmaf(hW1[h], s, suv);
    }
    const float m = sv * (1.f / HDIM);
    mv[d] = m;
    vv[d] = svv * (1.f / HDIM) - m * m;
    qv[d] = suv * (1.f / HDIM) - mu_u * m;
}

// Issue 16 coalesced async 512B copies: one contiguous 8 KB tile -> LDS.
// GVS mode: global = SGPR_base + voff + imm ; LDS = VGPR(ldsaddr) + imm.
__device__ __forceinline__ void async_tile_load(const float* __restrict__ g,
                                                unsigned ldsaddr, unsigned voff) {
#define PE_AL(OFF) asm volatile("global_load_async_to_lds_b128 %0, %1, %2 offset:" OFF \
        : : "v"(ldsaddr), "v"(voff), "s"(g) : "memory")
    PE_AL("0");    PE_AL("512");  PE_AL("1024"); PE_AL("1536");
    PE_AL("2048"); PE_AL("2560"); PE_AL("3072"); PE_AL("3584");
    PE_AL("4096"); PE_AL("4608"); PE_AL("5120"); PE_AL("5632");
    PE_AL("6144"); PE_AL("6656"); PE_AL("7168"); PE_AL("7680");
#undef PE_AL
}

// ---------------------------------------------------------------------------
// Kernel 2: main pipeline. Block = (b, d-chunk), 4 waves. Each wave owns
// 16-row tiles; the epg tile is streamed into a per-wave double-buffered LDS
// slab with global_load_async_to_lds_b128 (+ s_wait_asynccnt), overlapping the
// next tile's DMA with the current tile's WMMA work. Activations are built in
// the f32 WMMA A-layout; D = A @ h_W2 via V_WMMA_F32_16X16X4_F32 (K=128 -> 32
// steps x 2 N-tiles); LN2+ReLU via raw ds_swizzle xor-reductions (no generic
// shuffle overhead); deterministic per-block partials (no atomics).
// ---------------------------------------------------------------------------
__global__ void __launch_bounds__(BLOCK_T)
pe_main(const float* __restrict__ x, const int* __restrict__ mask,
        const float* __restrict__ epg, const float* __restrict__ mv,
        const float* __restrict__ qv, const float* __restrict__ vv,
        const float* __restrict__ scal,
        const float* __restrict__ hW1,
        const float* __restrict__ ln1g, const float* __restrict__ ln1b,
        const float* __restrict__ hW2, const float* __restrict__ hb2,
        const float* __restrict__ ln2g, const float* __restrict__ ln2b,
        float* __restrict__ pcsum, float* __restrict__ pmsum,
        int D, int nchunk) {
    __shared__ float  s_ep[NWAVES][2][TILE_FLOATS];   // 64 KB: per-wave double buffer
    __shared__ float2 s_w2p[(HDIM / 2) * CDIM];       // 16 KB: pre-swizzled B fragments
    __shared__ float  s_ug[HDIM], s_g[HDIM], s_c[HDIM];
    __shared__ float  s_b2[CDIM], s_g2[CDIM], s_c2[CDIM];
    __shared__ float  s_red[NWAVES][CDIM];
    __shared__ float  s_redm[NWAVES];

    const int tid = threadIdx.x;
    for (int i = tid; i < HDIM; i += BLOCK_T) {
        float gg = ln1g[i];
        s_ug[i] = hW1[i] * gg; s_g[i] = gg; s_c[i] = ln1b[i];
    }
    // B fragment (k0, k0+1) x (n) pre-swizzled: one ds_load_b64 per fragment.
    for (int i = tid; i < (HDIM / 2) * CDIM; i += BLOCK_T) {
        int p = i >> 5, n = i & 31;
        s_w2p[i] = make_float2(hW2[p * 64 + n], hW2[p * 64 + 32 + n]);
    }
    if (tid < CDIM) { s_b2[tid] = hb2[tid]; s_g2[tid] = ln2g[tid]; s_c2[tid] = ln2b[tid]; }
    __syncthreads();

    const int lane = tid & 31, wave = tid >> 5;
    const int hi = lane >> 4, lid = lane & 15;
    const int b = blockIdx.y;
    const int chunk = blockIdx.x;
    const int dbase = chunk * CHUNK_D;
    const int limit = min(TILES_PER_CHUNK, (D - dbase) / 16);  // valid tiles (D % 16 == 0)
    const float mu_u = scal[0], var_u = scal[1];

    const float b20 = s_b2[lid], b21 = s_b2[16 + lid];
    const float g20 = s_g2[lid], g21 = s_g2[16 + lid];
    const float c20 = s_c2[lid], c21 = s_c2[16 + lid];

    const unsigned voff   = (unsigned)(lane * 16);             // lane byte offset in tile
    const unsigned lbase0 = (unsigned)(unsigned long long)(const void*)&s_ep[wave][0][0];
    const unsigned lbase1 = (unsigned)(unsigned long long)(const void*)&s_ep[wave][1][0];
    const int bcast = hi << 5;    // bpermute byte base: lane (r + 8*hi) -> 4*r + 32*hi

    float cacc0 = 0.f, cacc1 = 0.f, mcacc = 0.f;

    // Prologue: DMA first tile into buffer 0.
    if (wave < limit)
        async_tile_load(epg + (size_t)(dbase + wave * 16) * HDIM, lbase0 + voff, voff);

    int iter = 0;
    for (int tile = wave; tile < limit; tile += NWAVES, ++iter) {
        const int nbuf = iter & 1;
        const bool havenext = (tile + NWAVES) < limit;
        if (havenext)   // prefetch next tile into the other buffer
            async_tile_load(epg + (size_t)(dbase + (tile + NWAVES) * 16) * HDIM,
                            (nbuf ? lbase0 : lbase1) + voff, voff);
        if (havenext) asm volatile("s_wait_asynccnt 0x10" ::: "memory");
        else          asm volatile("s_wait_asynccnt 0x0"  ::: "memory");

        const int drow = dbase + tile * 16 + lid;
        const float a  = x[(size_t)b * D + drow];
        const float mk = (float)mask[(size_t)b * D + drow];
        const float mstat = fmaf(a, mu_u, mv[drow]);
        const float var   = fmaf(a * a, var_u, fmaf(2.f * a, qv[drow], vv[drow]));
        const float inv   = rsqrtf(var + LN_EPS);
        const float negm  = -mstat;
        const float* tilep = &s_ep[wave][nbuf][lid * HDIM];

        f32x8 acc0 = {}; f32x8 acc1 = {};
        #pragma unroll 4
        for (int t = 0; t < HDIM / 4; ++t) {
            const int kp = 2 * t + hi;           // ISA A-layout: VGPR0 K={0|2}, VGPR1 K={1|3}
            const int k0 = 2 * kp;
            float2 e2 = *(const float2*)(tilep + k0);
            // elem = relu(((a*u + ep) - m)*inv*g + c) = relu((a*ug + epg - m*g)*inv + c)
            float f0 = fmaf(a, s_ug[k0],     e2.x); f0 = fmaf(negm, s_g[k0],     f0);
            float f1 = fmaf(a, s_ug[k0 + 1], e2.y); f1 = fmaf(negm, s_g[k0 + 1], f1);
            f32x2 af;
            af.x = fmaxf(fmaf(f0, inv, s_c[k0]),     0.f);
            af.y = fmaxf(fmaf(f1, inv, s_c[k0 + 1]), 0.f);
            float2 tb0 = s_w2p[kp * CDIM + lid];
            float2 tb1 = s_w2p[kp * CDIM + 16 + lid];
            f32x2 bf0; bf0.x = tb0.x; bf0.y = tb0.y;
            f32x2 bf1; bf1.x = tb1.x; bf1.y = tb1.y;
            acc0 = __builtin_amdgcn_wmma_f32_16x16x4_f32(false, af, false, bf0, (short)0, acc0, false, false);
            acc1 = __builtin_amdgcn_wmma_f32_16x16x4_f32(false, af, false, bf1, (short)0, acc1, false, false);
        }
        // Epilogue: +h_b2, LN over C=32 (lane-striped), ReLU, masked accumulate.
        // Reductions stay inside each 16-lane half (xor masks 1,2,4,8).
        #pragma unroll
        for (int r = 0; r < 8; ++r) {
            float o0 = acc0[r] + b20;
            float o1 = acc1[r] + b21;
            float s1 = o0 + o1;
            float s2 = fmaf(o0, o0, o1 * o1);
            s1 += SWZ_XOR(s1, 1);  s2 += SWZ_XOR(s2, 1);
            s1 += SWZ_XOR(s1, 2);  s2 += SWZ_XOR(s2, 2);
            s1 += SWZ_XOR(s1, 4);  s2 += SWZ_XOR(s2, 4);
            s1 += SWZ_XOR(s1, 8);  s2 += SWZ_XOR(s2, 8);
            const float mean = s1 * (1.f / CDIM);
            const float varr = fmaf(-mean, mean, s2 * (1.f / CDIM));
            const float i2 = rsqrtf(varr + LN_EPS);
            const float h0 = fmaxf(fmaf((o0 - mean) * i2, g20, c20), 0.f);
            const float h1 = fmaxf(fmaf((o1 - mean) * i2, g21, c21), 0.f);
            const float mrow = BPERM(bcast + 4 * r, mk);     // mask of row (r + 8*hi)
            cacc0 = fmaf(h0, mrow, cacc0);
            cacc1 = fmaf(h1, mrow, cacc1);
        }
        mcacc += mk;   // lanes 0-15 hold distinct rows; upper half duplicates
    }

    // Fold lane-halves (same columns); reduce mask count over lanes 0-15.
    cacc0 += SWZ_XOR(cacc0, 16);
    cacc1 += SWZ_XOR(cacc1, 16);
    mcacc += SWZ_XOR(mcacc, 1);
    mcacc += SWZ_XOR(mcacc, 2);
    mcacc += SWZ_XOR(mcacc, 4);
    mcacc += SWZ_XOR(mcacc, 8);

    if (lane < 16) { s_red[wave][lid] = cacc0; s_red[wave][16 + lid] = cacc1; }
    if (lane == 0) s_redm[wave] = mcacc;
    __syncthreads();
    if (wave == 0) {
        float s = 0.f;
        for (int w = 0; w < NWAVES; ++w) s += s_red[w][lane];
        pcsum[((size_t)b * nchunk + chunk) * CDIM + lane] = s;
        if (lane == 0) {
            float sm = 0.f;
            for (int w = 0; w < NWAVES; ++w) sm += s_redm[w];
            pmsum[(size_t)b * nchunk + chunk] = sm;
        }
    }
}

// ---------------------------------------------------------------------------
// Kernel 3: fixed-order chunk reduction (deterministic), masked mean, one-hot
// concat, then the tiny 36->128 (LN,ReLU) ->64 encoder head.
// ---------------------------------------------------------------------------
__global__ void pe_final(const float* __restrict__ pcs, const float* __restrict__ pms,
                         const int* __restrict__ cat,
                         const float* __restrict__ eW1, const float* __restrict__ eb1,
                         const float* __restrict__ eg, const float* __restrict__ ebn,
                         const float* __restrict__ eW2, const float* __restrict__ eb2,
                         float* __restrict__ out, int B, int nchunk) {
    __shared__ float cj[CDIM + NCAT];
    __shared__ float zpre[EDIM];
    __shared__ float zz[EDIM];
    const int b = blockIdx.x, t = threadIdx.x;
    if (t < CDIM) {
        float s = 0.f, sm = 0.f;
        for (int ch = 0; ch < nchunk; ++ch) {
            s  += pcs[((size_t)b * nchunk + ch) * CDIM + t];
            sm += pms[(size_t)b * nchunk + ch];
        }
        cj[t] = s / (sm + MEAN_EPS);
    } else if (t < CDIM + NCAT) {
        cj[t] = ((t - CDIM) == cat[b]) ? 1.f : 0.f;
    }
    __syncthreads();
    float s = eb1[t];
    #pragma unroll
    for (int j = 0; j < CDIM + NCAT; ++j) s = fmaf(cj[j], eW1[j * EDIM + t], s);
    zpre[t] = s;
    __syncthreads();
    float s1 = 0.f, s2 = 0.f;
    for (int i = 0; i < EDIM; ++i) { float v = zpre[i]; s1 += v; s2 = fmaf(v, v, s2); }
    const float mean = s1 * (1.f / EDIM);
    const float var  = s2 * (1.f / EDIM) - mean * mean;
    zz[t] = fmaxf(fmaf((s - mean) * rsqrtf(var + LN_EPS), eg[t], ebn[t]), 0.f);
    __syncthreads();
    if (t < 64) {
        float o = eb2[t];
        for (int e = 0; e < EDIM; ++e) o = fmaf(zz[e], eW2[e * 64 + t], o);
        if (t < 32) out[(size_t)b * 32 + t] = o;                         // mu
        else        out[(size_t)B * 32 + (size_t)b * 32 + (t - 32)] = o; // logvar
    }
}

extern "C" void kernel_launch(void* const* d_in, const int* in_sizes, int n_in,
                              void* d_out, int out_size, void* d_ws, size_t ws_size,
                              hipStream_t stream) {
    (void)n_in; (void)out_size; (void)ws_size;
    const float* x    = (const float*)d_in[0];
    const int*   mask = (const int*)  d_in[1];
    const int*   cat  = (const int*)  d_in[2];
    const float* fe   = (const float*)d_in[3];
    const float* hW1  = (const float*)d_in[4];
    const float* hb1  = (const float*)d_in[5];
    const float* ln1g = (const float*)d_in[6];
    const float* ln1b = (const float*)d_in[7];
    const float* hW2  = (const float*)d_in[8];
    const float* hb2  = (const float*)d_in[9];
    const float* ln2g = (const float*)d_in[10];
    const float* ln2b = (const float*)d_in[11];
    const float* eW1  = (const float*)d_in[12];
    const float* eb1  = (const float*)d_in[13];
    const float* eg   = (const float*)d_in[14];
    const float* ebn  = (const float*)d_in[15];
    const float* eW2  = (const float*)d_in[16];
    const float* eb2  = (const float*)d_in[17];
    float* out = (float*)d_out;

    const int B = in_sizes[2];
    const int D = in_sizes[0] / B;
    const int nchunk = (D + CHUNK_D - 1) / CHUNK_D;

    // workspace layout (floats): epg | mv | qv | vv | scal(16) | pcsum | pmsum (~10.7 MB)
    float* ws   = (float*)d_ws;
    float* epg  = ws;
    float* mv   = epg + (size_t)D * HDIM;
    float* qv   = mv + D;
    float* vv   = qv + D;
    float* scal = vv + D;
    float* pcs  = scal + 16;
    float* pms  = pcs + (size_t)B * nchunk * CDIM;

    pe_precompute<<<(D + 127) / 128, 128, 0, stream>>>(fe, hW1, hb1, ln1g,
                                                       epg, mv, qv, vv, scal, D);
    dim3 grid(nchunk, B);
    pe_main<<<grid, BLOCK_T, 0, stream>>>(x, mask, epg, mv, qv, vv, scal, hW1, ln1g, ln1b,
                                          hW2, hb2, ln2g, ln2b, pcs, pms, D, nchunk);
    pe_final<<<B, EDIM, 0, stream>>>(pcs, pms, cat, eW1, eb1, eg, ebn, eW2, eb2, out, B, nchunk);
}